// IMPGNN_60911226191914
// MI455X (gfx1250) — compile-verified
//
#include <hip/hip_runtime.h>
#include <hip/hip_bf16.h>

#define HDIM 128
#define H4 (HDIM / 4)

typedef float v2f __attribute__((ext_vector_type(2)));
typedef float v8f __attribute__((ext_vector_type(8)));

__device__ __forceinline__ float elu1(float x) {
    return x > 0.f ? x : (__expf(x) - 1.f);
}

// ---------------------------------------------------------------- utilities
__global__ void k_zero(float* __restrict__ p, long n) {
    long i = (long)blockIdx.x * blockDim.x + threadIdx.x;
    if (i < n) p[i] = 0.f;
}

__global__ void k_copy(float* __restrict__ d, const float* __restrict__ s, long n) {
    long i = (long)blockIdx.x * blockDim.x + threadIdx.x;
    if (i < n) d[i] = s[i];
}

// ------------------------------------------------------- embedding gathers
__global__ void k_gather_rows(const float* __restrict__ emb, const int* __restrict__ idx,
                              float* __restrict__ out, long rows) {
    long i = (long)blockIdx.x * blockDim.x + threadIdx.x;   // rows * H/4
    long total = rows * H4;
    if (i >= total) return;
    long r = i / H4;
    int c4 = (int)(i % H4);
    const float4* s = (const float4*)(emb + (long)idx[r] * HDIM);
    ((float4*)(out + r * HDIM))[c4] = s[c4];
}

// --------------------------------------------------------- metafeat encoder
__global__ void k_metafeat(const float* __restrict__ mfeat, const float* __restrict__ Wm,
                           const float* __restrict__ bm, float* __restrict__ out,
                           long n, int npath) {
    long i = (long)blockIdx.x * blockDim.x + threadIdx.x;
    if (i >= n * HDIM) return;
    long r = i / HDIM;
    int h = (int)(i % HDIM);
    float acc = bm[h];
    for (int p = 0; p < npath; ++p)
        acc = fmaf(mfeat[r * npath + p], Wm[p * HDIM + h], acc);
    out[i] = acc;
}

// --------------------------- T @ ef : out[src]+=v, out[dst]+=v (per 4 cols)
__global__ void k_scatter_both(const float* __restrict__ v, const int* __restrict__ src,
                               const int* __restrict__ dst, float* __restrict__ out, long e) {
    long i = (long)blockIdx.x * blockDim.x + threadIdx.x;   // e * H/4
    long total = e * H4;
    if (i >= total) return;
    long ed = i / H4;
    int c = (int)(i % H4) * 4;
    float4 val = *(const float4*)(v + ed * HDIM + c);
    float* ps = out + (long)src[ed] * HDIM + c;
    float* pd = out + (long)dst[ed] * HDIM + c;
    atomicAdd(ps + 0, val.x); atomicAdd(ps + 1, val.y);
    atomicAdd(ps + 2, val.z); atomicAdd(ps + 3, val.w);
    atomicAdd(pd + 0, val.x); atomicAdd(pd + 1, val.y);
    atomicAdd(pd + 2, val.z); atomicAdd(pd + 3, val.w);
}

// --------------------------- two gated ELUs sharing one gate read
__global__ void k_mul_elu2(const float* __restrict__ a, const float* __restrict__ b,
                           const float* __restrict__ g,
                           float* __restrict__ oa, float* __restrict__ ob, long n) {
    long i = (long)blockIdx.x * blockDim.x + threadIdx.x;
    if (i >= n) return;
    float gg = g[i];
    oa[i] = elu1(a[i] * gg);
    ob[i] = elu1(b[i] * gg);
}

// --------------------------- ef = elu(ef0 * Tt(nf0) * Tt(mf0))
__global__ void k_edge_elu(const float* __restrict__ ef0, const float* __restrict__ nf0,
                           const float* __restrict__ mf0, const int* __restrict__ src,
                           const int* __restrict__ dst, float* __restrict__ out, long e) {
    long i = (long)blockIdx.x * blockDim.x + threadIdx.x;
    long total = e * HDIM;
    if (i >= total) return;
    long ed = i / HDIM;
    int h = (int)(i % HDIM);
    long s = (long)src[ed] * HDIM + h;
    long d = (long)dst[ed] * HDIM + h;
    out[i] = elu1(ef0[i] * (nf0[s] + nf0[d]) * (mf0[s] + mf0[d]));
}

// --------------------------- degree count (raw, no self-loop)
__global__ void k_deg(const int* __restrict__ src, const int* __restrict__ dst,
                      float* __restrict__ deg, long e) {
    long i = (long)blockIdx.x * blockDim.x + threadIdx.x;
    if (i >= e) return;
    atomicAdd(&deg[src[i]], 1.f);
    atomicAdd(&deg[dst[i]], 1.f);
}

// --------------------------------------------- WMMA f32 GEMM: Y = X @ W
// X: M x 128, W: 128 x 128, Y: M x 128.  M must be a multiple of 16.
// Block = 256 threads = 8 wave32; wave w owns output columns [16w, 16w+16).
__global__ __launch_bounds__(256) void k_gemm_wmma(const float* __restrict__ X,
                                                   const float* __restrict__ W,
                                                   float* __restrict__ Y, long M) {
    __shared__ float As[16 * 132];               // 16 rows, stride 132 (bank-pad)
    const int tid  = threadIdx.x;
    const int lane = tid & 31;
    const int wave = tid >> 5;
    const long row0 = (long)blockIdx.x * 16;

    // stage 16x128 A tile into LDS: thread t loads row t>>4, 8 cols
    {
        const int r = tid >> 4;
        const int c = (tid & 15) * 8;
        const float4* g = (const float4*)(X + (row0 + r) * HDIM + c);
        float4* s = (float4*)(As + r * 132 + c);
        s[0] = g[0];
        s[1] = g[1];
    }
    __syncthreads();

    const int n0   = wave * 16;
    const int mrow = lane & 15;
    const int koff = (lane >> 4) * 2;            // lanes 0-15: K 0,1 ; 16-31: K 2,3
    v8f c = {};
#pragma unroll
    for (int k = 0; k < HDIM; k += 4) {
        v2f a, b;
        a.x = As[mrow * 132 + k + koff];
        a.y = As[mrow * 132 + k + koff + 1];
        b.x = W[(k + koff) * HDIM + n0 + mrow];
        b.y = W[(k + koff + 1) * HDIM + n0 + mrow];
        c = __builtin_amdgcn_wmma_f32_16x16x4_f32(false, a, false, b,
                                                  (short)0, c, false, false);
    }
    const int mbase = (lane >> 4) * 8;           // C/D: VGPR r -> M=r (lo) / M=r+8 (hi)
#pragma unroll
    for (int r = 0; r < 8; ++r)
        Y[(row0 + mbase + r) * HDIM + n0 + mrow] = c[r];
}

// --------------------------- m = dinv * xw   (dinv = rsqrt(deg+1), GCN self-loop)
__global__ void k_scale_dinv(float* __restrict__ x, const float* __restrict__ deg, long rows) {
    long i = (long)blockIdx.x * blockDim.x + threadIdx.x;
    if (i >= rows * HDIM) return;
    long r = i / HDIM;
    x[i] *= rsqrtf(deg[r] + 1.f);
}

// --------------------------- agg[dst]+=m[src]; agg[src]+=m[dst]
__global__ void k_gcn_scatter(const float* __restrict__ m, const int* __restrict__ src,
                              const int* __restrict__ dst, float* __restrict__ out, long e) {
    long i = (long)blockIdx.x * blockDim.x + threadIdx.x;   // e * H/4
    long total = e * H4;
    if (i >= total) return;
    long ed = i / H4;
    int c = (int)(i % H4) * 4;
    long s = (long)src[ed], d = (long)dst[ed];
    float4 vs = *(const float4*)(m + s * HDIM + c);
    float4 vd = *(const float4*)(m + d * HDIM + c);
    float* pd = out + d * HDIM + c;
    float* ps = out + s * HDIM + c;
    atomicAdd(pd + 0, vs.x); atomicAdd(pd + 1, vs.y);
    atomicAdd(pd + 2, vs.z); atomicAdd(pd + 3, vs.w);
    atomicAdd(ps + 0, vd.x); atomicAdd(ps + 1, vd.y);
    atomicAdd(ps + 2, vd.z); atomicAdd(ps + 3, vd.w);
}

// --------------------------- out = dinv*agg + bias
__global__ void k_gcn_finish(float* __restrict__ out, const float* __restrict__ deg,
                             const float* __restrict__ bias, long rows) {
    long i = (long)blockIdx.x * blockDim.x + threadIdx.x;
    if (i >= rows * HDIM) return;
    long r = i / HDIM;
    int h = (int)(i % HDIM);
    out[i] = out[i] * rsqrtf(deg[r] + 1.f) + bias[h];
}

// ---------------------------------------------------------------- pagerank
__global__ void k_pr_init(float* __restrict__ pr, long n) {
    long i = (long)blockIdx.x * blockDim.x + threadIdx.x;
    if (i < n) pr[i] = 1.f / (float)n;
}

__global__ void k_pr_prepare(const float* __restrict__ pr, const float* __restrict__ deg,
                             float* __restrict__ c, float* __restrict__ agg, long n) {
    long i = (long)blockIdx.x * blockDim.x + threadIdx.x;
    if (i >= n) return;
    c[i] = pr[i] / fmaxf(deg[i], 1.f);
    agg[i] = 0.f;
}

__global__ void k_pr_scatter(const float* __restrict__ c, const int* __restrict__ src,
                             const int* __restrict__ dst, float* __restrict__ agg, long e) {
    long i = (long)blockIdx.x * blockDim.x + threadIdx.x;
    if (i >= e) return;
    atomicAdd(&agg[dst[i]], c[src[i]]);
    atomicAdd(&agg[src[i]], c[dst[i]]);
}

__global__ void k_pr_update(const float* __restrict__ agg, float* __restrict__ pr, long n) {
    long i = (long)blockIdx.x * blockDim.x + threadIdx.x;
    if (i >= n) return;
    pr[i] = 0.15f / (float)n + 0.85f * agg[i];
}

// --------------------------- out = a * pr[row]   (writes into d_out)
__global__ void k_scale_pr(const float* __restrict__ a, const float* __restrict__ pr,
                           float* __restrict__ out, long rows) {
    long i = (long)blockIdx.x * blockDim.x + threadIdx.x;
    if (i >= rows * HDIM) return;
    long r = i / HDIM;
    out[i] = a[i] * pr[r];
}

// --------------------------- column sum pooling (128 threads, 256-row stripes)
__global__ void k_colsum(const float* __restrict__ A, float* __restrict__ z, long rows) {
    int col = threadIdx.x;                         // 0..127
    long r0 = (long)blockIdx.x * 256;
    long r1 = r0 + 256; if (r1 > rows) r1 = rows;
    float acc = 0.f;
    for (long r = r0; r < r1; ++r) acc += A[r * HDIM + col];
    atomicAdd(&z[col], acc);
}

// --------------------------- head: Z = zcat @ cat2_W + b ; pred = Z @ pred_W + b
__global__ void k_head(const float* __restrict__ z, const float* __restrict__ cat2_W,
                       const float* __restrict__ cat2_b, const float* __restrict__ pred_W,
                       const float* __restrict__ pred_b, float* __restrict__ pred, int task) {
    __shared__ float Z[HDIM];
    int h = threadIdx.x;                           // 128 threads
    float acc = cat2_b[h];
    for (int k = 0; k < 2 * HDIM; ++k)
        acc = fmaf(z[k], cat2_W[k * HDIM + h], acc);
    Z[h] = acc;
    __syncthreads();
    if (h < task) {
        float p = pred_b[h];
        for (int j = 0; j < HDIM; ++j)
            p = fmaf(Z[j], pred_W[j * task + h], p);
        pred[h] = p;
    }
}

// ===========================================================================
extern "C" void kernel_launch(void* const* d_in, const int* in_sizes, int n_in,
                              void* d_out, int out_size, void* d_ws, size_t ws_size,
                              hipStream_t stream) {
    const int*   x          = (const int*)d_in[0];
    const float* metafeat   = (const float*)d_in[1];
    const int*   edge_attr  = (const int*)d_in[2];
    const int*   edge_index = (const int*)d_in[3];
    const int*   lg_ei      = (const int*)d_in[4];
    const float* atom_emb   = (const float*)d_in[5];
    const float* bond_emb   = (const float*)d_in[6];
    const float* meta_W     = (const float*)d_in[7];
    const float* meta_b     = (const float*)d_in[8];
    const float* org_W      = (const float*)d_in[9];
    const float* org_b      = (const float*)d_in[10];
    const float* org1_W     = (const float*)d_in[11];
    const float* org1_b     = (const float*)d_in[12];
    const float* mconv_W    = (const float*)d_in[13];
    const float* mconv_b    = (const float*)d_in[14];
    const float* mconv1_W   = (const float*)d_in[15];
    const float* mconv1_b   = (const float*)d_in[16];
    const float* lg_W       = (const float*)d_in[17];
    const float* lg_b       = (const float*)d_in[18];
    // d_in[19], d_in[20] (lg1_W/lg1_b): result unused downstream — skipped.
    const float* cat2_W     = (const float*)d_in[21];
    const float* cat2_b     = (const float*)d_in[22];
    const float* pred_W     = (const float*)d_in[23];
    const float* pred_b     = (const float*)d_in[24];

    const long N     = in_sizes[0];
    const long E     = in_sizes[2];
    const long ELG   = in_sizes[4] / 2;
    const int  NPATH = (int)(in_sizes[1] / N);
    const int  TASK  = in_sizes[24];

    const int* src = edge_index;
    const int* dst = edge_index + E;
    const int* ls  = lg_ei;
    const int* ld  = lg_ei + ELG;

    const long NH = N * HDIM, EH = E * HDIM;

    float* w = (float*)d_ws;
    float* nf0    = w;  w += NH;
    float* mf0    = w;  w += NH;
    float* gate   = w;  w += NH;
    float* nf     = w;  w += NH;
    float* mf     = w;  w += NH;
    float* h_org  = w;  w += NH;
    float* h_meta = w;  w += NH;
    float* ebufA  = w;  w += EH;     // ef0 -> m (line graph)
    float* ebufB  = w;  w += EH;     // ef  -> h_lg
    float* deg_n  = w;  w += N;
    float* deg_e  = w;  w += E;
    float* pr     = w;  w += N;
    float* prc    = w;  w += N;
    float* pragg  = w;  w += N;
    float* zpool  = w;  w += 2 * HDIM;

    float* pred     = (float*)d_out;
    float* out_meta = pred + TASK;
    float* out_org  = out_meta + NH;

    auto nb = [](long n) { return dim3((unsigned)((n + 255) / 256)); };

    // ---- encoders
    k_gather_rows<<<nb(N * H4), 256, 0, stream>>>(atom_emb, x, nf0, N);
    k_metafeat<<<nb(NH), 256, 0, stream>>>(metafeat, meta_W, meta_b, mf0, N, NPATH);
    k_gather_rows<<<nb(E * H4), 256, 0, stream>>>(bond_emb, edge_attr, ebufA, E);

    // ---- gate_n = T @ ef0 ; nf, mf, ef
    k_zero<<<nb(NH), 256, 0, stream>>>(gate, NH);
    k_scatter_both<<<nb(E * H4), 256, 0, stream>>>(ebufA, src, dst, gate, E);
    k_mul_elu2<<<nb(NH), 256, 0, stream>>>(nf0, mf0, gate, nf, mf, NH);
    k_edge_elu<<<nb(EH), 256, 0, stream>>>(ebufA, nf0, mf0, src, dst, ebufB, E);

    // ---- degrees
    k_zero<<<nb(N), 256, 0, stream>>>(deg_n, N);
    k_deg<<<nb(E), 256, 0, stream>>>(src, dst, deg_n, E);
    k_zero<<<nb(E), 256, 0, stream>>>(deg_e, E);
    k_deg<<<nb(ELG), 256, 0, stream>>>(ls, ld, deg_e, ELG);

    // ---- GCN layer 1 (nodes): h_org, h_meta
    k_gemm_wmma<<<dim3((unsigned)(N / 16)), 256, 0, stream>>>(nf, org_W, nf0, N);
    k_scale_dinv<<<nb(NH), 256, 0, stream>>>(nf0, deg_n, N);
    k_copy<<<nb(NH), 256, 0, stream>>>(h_org, nf0, NH);
    k_gcn_scatter<<<nb(E * H4), 256, 0, stream>>>(nf0, src, dst, h_org, E);
    k_gcn_finish<<<nb(NH), 256, 0, stream>>>(h_org, deg_n, org_b, N);

    k_gemm_wmma<<<dim3((unsigned)(N / 16)), 256, 0, stream>>>(mf, mconv_W, mf0, N);
    k_scale_dinv<<<nb(NH), 256, 0, stream>>>(mf0, deg_n, N);
    k_copy<<<nb(NH), 256, 0, stream>>>(h_meta, mf0, NH);
    k_gcn_scatter<<<nb(E * H4), 256, 0, stream>>>(mf0, src, dst, h_meta, E);
    k_gcn_finish<<<nb(NH), 256, 0, stream>>>(h_meta, deg_n, mconv_b, N);

    // ---- GCN (line graph): h_lg in ebufB
    k_gemm_wmma<<<dim3((unsigned)(E / 16)), 256, 0, stream>>>(ebufB, lg_W, ebufA, E);
    k_scale_dinv<<<nb(EH), 256, 0, stream>>>(ebufA, deg_e, E);
    k_copy<<<nb(EH), 256, 0, stream>>>(ebufB, ebufA, EH);
    k_gcn_scatter<<<nb(ELG * H4), 256, 0, stream>>>(ebufA, ls, ld, ebufB, ELG);
    k_gcn_finish<<<nb(EH), 256, 0, stream>>>(ebufB, deg_e, lg_b, E);

    // ---- gate_l = T @ h_lg ; hats
    k_zero<<<nb(NH), 256, 0, stream>>>(gate, NH);
    k_scatter_both<<<nb(E * H4), 256, 0, stream>>>(ebufB, src, dst, gate, E);
    k_mul_elu2<<<nb(NH), 256, 0, stream>>>(h_org, h_meta, gate, nf, mf, NH);

    // ---- GCN layer 2 (nodes): overwrite h_org, h_meta  (lg1 branch is dead code)
    k_gemm_wmma<<<dim3((unsigned)(N / 16)), 256, 0, stream>>>(nf, org1_W, nf0, N);
    k_scale_dinv<<<nb(NH), 256, 0, stream>>>(nf0, deg_n, N);
    k_copy<<<nb(NH), 256, 0, stream>>>(h_org, nf0, NH);
    k_gcn_scatter<<<nb(E * H4), 256, 0, stream>>>(nf0, src, dst, h_org, E);
    k_gcn_finish<<<nb(NH), 256, 0, stream>>>(h_org, deg_n, org1_b, N);

    k_gemm_wmma<<<dim3((unsigned)(N / 16)), 256, 0, stream>>>(mf, mconv1_W, mf0, N);
    k_scale_dinv<<<nb(NH), 256, 0, stream>>>(mf0, deg_n, N);
    k_copy<<<nb(NH), 256, 0, stream>>>(h_meta, mf0, NH);
    k_gcn_scatter<<<nb(E * H4), 256, 0, stream>>>(mf0, src, dst, h_meta, E);
    k_gcn_finish<<<nb(NH), 256, 0, stream>>>(h_meta, deg_n, mconv1_b, N);

    // ---- pagerank (100 iters)
    k_pr_init<<<nb(N), 256, 0, stream>>>(pr, N);
    for (int it = 0; it < 100; ++it) {
        k_pr_prepare<<<nb(N), 256, 0, stream>>>(pr, deg_n, prc, pragg, N);
        k_pr_scatter<<<nb(E), 256, 0, stream>>>(prc, src, dst, pragg, E);
        k_pr_update<<<nb(N), 256, 0, stream>>>(pragg, pr, N);
    }

    // ---- scale by pagerank, write outputs
    k_scale_pr<<<nb(NH), 256, 0, stream>>>(h_meta, pr, out_meta, N);
    k_scale_pr<<<nb(NH), 256, 0, stream>>>(h_org, pr, out_org, N);

    // ---- pooling + head
    k_zero<<<nb(2 * HDIM), 256, 0, stream>>>(zpool, 2 * HDIM);
    k_colsum<<<nb(N * 256 / 256 > 0 ? (N + 255) / 256 : 1) , 128, 0, stream>>>(out_meta, zpool, N);
    k_colsum<<<dim3((unsigned)((N + 255) / 256)), 128, 0, stream>>>(out_org, zpool + HDIM, N);
    k_head<<<1, HDIM, 0, stream>>>(zpool, cat2_W, cat2_b, pred_W, pred_b, pred, TASK);
}